// RFLAF_Poly_88029649699201
// MI455X (gfx1250) — compile-verified
//
#include <hip/hip_runtime.h>

#define BATCH   8192
#define KDIM    1024
#define HDIM    2048
#define NCOEF   401
#define NSPLIT  4                          // column splits over blockIdx.y
#define COLS_PER_SPLIT (HDIM / NSPLIT)     // 512
#define WAVES   4
#define TPB     (WAVES * 32)               // 128 threads = 4 wave32

// LDS pads tuned for the WMMA half-wave access pattern (see analysis):
#define XS_STRIDE 36   // 16B-aligned rows; 36r mod 64 distinct; hi-half +2 never aliases
#define WS_STRIDE 72   // bank = 8*row + n: hi half (rows K+2/3) lands 16 banks away

typedef float v2f __attribute__((ext_vector_type(2)));
typedef float v8f __attribute__((ext_vector_type(8)));

#if defined(__has_builtin)
#if __has_builtin(__builtin_amdgcn_global_load_async_to_lds_b128)
#define HAVE_ASYNC_LDS 1
#endif
#endif

#if HAVE_ASYNC_LDS
// Builtin param types (from clang diagnostic): pointer to GNU-vector int4,
// AS1 (printed "__device__") for the global source, AS3 for the LDS dest.
typedef int gnu_v4i __attribute__((vector_size(4 * sizeof(int))));
typedef __attribute__((address_space(1))) gnu_v4i as1_v4i;
typedef __attribute__((address_space(3))) gnu_v4i as3_v4i;

__device__ __forceinline__ void async_b128(const float* g, float* l) {
  // GLOBAL_LOAD_ASYNC_TO_LDS_B128: 16B per lane, memory -> LDS, ASYNCcnt-tracked
  __builtin_amdgcn_global_load_async_to_lds_b128(
      (as1_v4i*)g, (as3_v4i*)l, /*imm offset=*/0, /*cpol=*/0);
}
#if __has_builtin(__builtin_amdgcn_s_wait_asynccnt)
#define WAIT_ASYNC(N) __builtin_amdgcn_s_wait_asynccnt(N)
#else
#define WAIT_ASYNC(N) asm volatile("s_wait_asynccnt %0" ::"i"(N) : "memory")
#endif
#endif

// Fused: C_tile = x@W via V_WMMA_F32_16X16X4_F32, Horner(a, C) on accumulator
// registers, scale by v[col], column-reduce -> per-row partials in d_ws.
__global__ __launch_bounds__(TPB) void rflaf_fused(
    const float* __restrict__ x, const float* __restrict__ W,
    const float* __restrict__ v, const float* __restrict__ a,
    float* __restrict__ part)
{
  __shared__ alignas(16) float Xs[2][16][XS_STRIDE];
  __shared__ alignas(16) float Ws[2][32][WS_STRIDE];
  __shared__ float As[NCOEF];
  __shared__ float Red[WAVES][16];

  const int tid  = threadIdx.x;
  const int wave = tid >> 5;
  const int lane = tid & 31;
  const int hi   = lane >> 4;      // half-wave: K/M offset in WMMA fragment layout
  const int l16  = lane & 15;

  const int rowBase = blockIdx.x << 4;               // 16 rows per block
  const int colBase = blockIdx.y * COLS_PER_SPLIT;   // 512-col slice

  // per-thread staging coordinates (coalesced 16B per lane)
  const int xr = tid >> 3;          // 0..15  (x tile row)
  const int xc = (tid & 7) << 2;    // 0..28  (x tile k, float4)
  const int wr = tid >> 4;          // 0..7   (W tile row within pass)
  const int wc = (tid & 15) << 2;   // 0..60  (W tile col, float4)

  for (int i = tid; i < NCOEF; i += TPB) As[i] = a[i];

  float acc[8];
  #pragma unroll
  for (int i = 0; i < 8; ++i) acc[i] = 0.0f;

  for (int it = 0; it < COLS_PER_SPLIT / 64; ++it) {
    const int c0 = colBase + it * 64;                // this wave: cols c0+wave*16..
    v8f t = {};                                      // 16x16 f32 accumulator (8 VGPRs)

#if HAVE_ASYNC_LDS
    // prime the pipeline: chunk 0 -> buffer 0 (5 async transfers per wave)
    async_b128(&x[(rowBase + xr) * KDIM + xc], &Xs[0][xr][xc]);
    #pragma unroll
    for (int p = 0; p < 4; ++p)
      async_b128(&W[(p * 8 + wr) * HDIM + c0 + wc], &Ws[0][p * 8 + wr][wc]);
#endif

    for (int k0 = 0; k0 < KDIM; k0 += 32) {
#if HAVE_ASYNC_LDS
      const int buf = (k0 >> 5) & 1;
      // all waves done reading buf^1 (previous chunk) before overwrite
      __syncthreads();
      if (k0 + 32 < KDIM) {
        const int nk = k0 + 32;
        async_b128(&x[(rowBase + xr) * KDIM + nk + xc], &Xs[buf ^ 1][xr][xc]);
        #pragma unroll
        for (int p = 0; p < 4; ++p)
          async_b128(&W[(nk + p * 8 + wr) * HDIM + c0 + wc],
                     &Ws[buf ^ 1][p * 8 + wr][wc]);
        WAIT_ASYNC(5);   // async loads complete in order: <=5 left => chunk k0 landed
      } else {
        WAIT_ASYNC(0);
      }
      __syncthreads();
#else
      const int buf = 0;
      __syncthreads();
      *reinterpret_cast<float4*>(&Xs[0][xr][xc]) =
          *reinterpret_cast<const float4*>(&x[(rowBase + xr) * KDIM + k0 + xc]);
      #pragma unroll
      for (int p = 0; p < 4; ++p)
        *reinterpret_cast<float4*>(&Ws[0][p * 8 + wr][wc]) =
            *reinterpret_cast<const float4*>(&W[(k0 + p * 8 + wr) * HDIM + c0 + wc]);
      if (k0 + 32 < KDIM) {
        __builtin_prefetch(&x[(rowBase + xr) * KDIM + k0 + 32], 0, 1);
        __builtin_prefetch(&W[(k0 + 32 + wr) * HDIM + c0], 0, 1);
      }
      __syncthreads();
#endif

      // 8 WMMA steps of K=4: A 16x4 (VGPR0: K={0|2}, VGPR1: K={1|3} per half)
      #pragma unroll
      for (int kk = 0; kk < 32; kk += 4) {
        v2f Af, Bf;
        Af.x = Xs[buf][l16][kk + 2 * hi];
        Af.y = Xs[buf][l16][kk + 2 * hi + 1];
        const int n = (wave << 4) + l16;
        Bf.x = Ws[buf][kk + 2 * hi][n];
        Bf.y = Ws[buf][kk + 2 * hi + 1][n];
        t = __builtin_amdgcn_wmma_f32_16x16x4_f32(false, Af, false, Bf,
                                                  (short)0, t, false, false);
      }
    }

    // Horner: res = res*xW + a[n], n = 400..0, on the accumulator registers
    v8f p = {};
    #pragma unroll 4
    for (int n = NCOEF - 1; n >= 0; --n) {
      const float coef = As[n];
      #pragma unroll
      for (int i = 0; i < 8; ++i) p[i] = __builtin_fmaf(p[i], t[i], coef);
    }

    // scale by v[col]; butterfly-reduce the 16 columns within each half-wave
    const float vv = v[c0 + (wave << 4) + l16];
    #pragma unroll
    for (int i = 0; i < 8; ++i) {
      float r = p[i] * vv;
      r += __shfl_xor(r, 1, 32);
      r += __shfl_xor(r, 2, 32);
      r += __shfl_xor(r, 4, 32);
      r += __shfl_xor(r, 8, 32);
      acc[i] += r;   // lane 0 -> rows rowBase+i, lane 16 -> rows rowBase+8+i
    }
  }

  // deterministic cross-wave reduction (no float atomics)
  __syncthreads();
  if (l16 == 0) {
    #pragma unroll
    for (int i = 0; i < 8; ++i) Red[wave][hi * 8 + i] = acc[i];
  }
  __syncthreads();
  if (tid < 16) {
    const float s = (Red[0][tid] + Red[1][tid]) + (Red[2][tid] + Red[3][tid]);
    part[blockIdx.y * BATCH + rowBase + tid] = s;
  }
}

__global__ void rflaf_reduce(const float* __restrict__ part,
                             float* __restrict__ out) {
  const int i = blockIdx.x * blockDim.x + threadIdx.x;
  if (i < BATCH)
    out[i] = (part[i] + part[BATCH + i]) + (part[2 * BATCH + i] + part[3 * BATCH + i]);
}

extern "C" void kernel_launch(void* const* d_in, const int* /*in_sizes*/, int /*n_in*/,
                              void* d_out, int /*out_size*/, void* d_ws, size_t /*ws_size*/,
                              hipStream_t stream) {
  const float* x = (const float*)d_in[0];
  const float* W = (const float*)d_in[1];
  const float* v = (const float*)d_in[2];
  const float* a = (const float*)d_in[3];
  float* part = (float*)d_ws;   // NSPLIT * 8192 floats = 128 KB scratch

  dim3 grid(BATCH / 16, NSPLIT);
  rflaf_fused<<<grid, TPB, 0, stream>>>(x, W, v, a, part);
  rflaf_reduce<<<(BATCH + 255) / 256, 256, 0, stream>>>(part, (float*)d_out);
}